// InstanceDiceLoss_86517821211941
// MI455X (gfx1250) — compile-verified
//
#include <hip/hip_runtime.h>

// ---------------- problem geometry (fixed by the reference) ----------------
#define NIMG   8
#define HDIM   960
#define WDIM   960
#define HW     (HDIM * WDIM)      // 921600 pixels per image
#define IOU_T  0.2f
#define EPS_F  1e-6f

// main kernel decomposition: 90 blocks/image * 8 waves * 32 lanes
//   -> 720 waves per image, 28800 chunks of 32 px, exactly 40 chunks/wave.
#define BLOCKS_PER_IMG 90
#define THREADS        256
#define WAVES_PER_BLK  8

// scratch layout: partials[img][q][16][16], q in {0:ones, 1:A=p*g*m,
// 2:Bp=p*m, 3:Bg=g*m}.  Each 16x16 matrix D holds (wildcard scheme):
//   D[k][j], k,j<15 : pair sum for gt label k+1, pred label j+1
//   D[15][j]        : by_p marginal for pred label j+1 (all gt labels 0..16)
//   D[k][15]        : by_g marginal for gt  label k+1 (all pred labels 0..16)
#define MATS      4
#define MAT_ELEMS 256
#define ACC_PER_IMG (MATS * MAT_ELEMS)

typedef float v2f __attribute__((ext_vector_type(2)));
typedef float v8f __attribute__((ext_vector_type(8)));

static __device__ __forceinline__ v8f wmma_f32_k4(v2f a, v2f b, v8f c) {
    // D = A(16x4,f32) * B(4x16,f32) + C(16x16,f32)
    return __builtin_amdgcn_wmma_f32_16x16x4_f32(
        /*neg_a=*/false, a, /*neg_b=*/false, b,
        /*c_mod=*/(short)0, c, /*reuse_a=*/false, /*reuse_b=*/false);
}

// ---------------------------------------------------------------------------
__global__ void zero_ws_kernel(float* __restrict__ p, int n) {
    int i = blockIdx.x * blockDim.x + threadIdx.x;
    if (i < n) p[i] = 0.0f;
}

// ---------------------------------------------------------------------------
// One-hot (wildcard-row) WMMA segment-sum over pixels.
__global__ __launch_bounds__(THREADS)
void seg_wmma_kernel(const float* __restrict__ pred,
                     const float* __restrict__ gt,
                     const float* __restrict__ mask,
                     const int*   __restrict__ plab,
                     const int*   __restrict__ glab,
                     float* __restrict__ partials) {
    __shared__ float lds[ACC_PER_IMG];

    const int img  = blockIdx.x / BLOCKS_PER_IMG;
    const int blk  = blockIdx.x % BLOCKS_PER_IMG;
    const int wave = threadIdx.x >> 5;
    const int lane = threadIdx.x & 31;

    for (int t = threadIdx.x; t < ACC_PER_IMG; t += THREADS) lds[t] = 0.0f;
    __syncthreads();

    v8f acc[MATS];
#pragma unroll
    for (int q = 0; q < MATS; ++q) {
        v8f z = {0.f, 0.f, 0.f, 0.f, 0.f, 0.f, 0.f, 0.f};
        acc[q] = z;
    }

    // uniform per-image base pointers -> SADDR + 32-bit voffset addressing
    const long ib = (long)img * HW;
    const float* __restrict__ pP = pred + ib;
    const float* __restrict__ pG = gt   + ib;
    const float* __restrict__ pM = mask + ib;
    const int*   __restrict__ pL = plab + ib;
    const int*   __restrict__ gL = glab + ib;

    const int   row    = lane & 15;          // A row / B col owned by this lane
    const int   tgt    = row + 1;            // label mapped to this row/col
    const float lastF  = (row == 15) ? 1.0f : 0.0f;   // wildcard row/col
    const int   khalf  = (lane >> 4) << 1;   // lanes 0-15 own K={0,1}, 16-31 K={2,3}
    const int   stride = BLOCKS_PER_IMG * WAVES_PER_BLK * 32;

    for (int base = (blk * WAVES_PER_BLK + wave) * 32; base + 32 <= HW;
         base += stride) {
        const int idx = base + lane;
        const float p  = __builtin_nontemporal_load(&pP[idx]);
        const float gv = __builtin_nontemporal_load(&pG[idx]);
        const float mv = __builtin_nontemporal_load(&pM[idx]);
        const int   pl = __builtin_nontemporal_load(&pL[idx]);
        const int   gl = __builtin_nontemporal_load(&gL[idx]);
        const float xa  = p * gv * mv;            // pred*gt*mask
        const float xbp = p * mv;                 // pred*mask
        const float xbg = gv * mv;                // gt*mask
        const int   lab = gl | (pl << 16);        // both labels in one dword

#pragma unroll
        for (int kk = 0; kk < 32; kk += 4) {
            const int k0 = kk + khalf;
            const int k1 = k0 + 1;
            // broadcast pixel k0/k1 data to every lane (ds_bpermute)
            const int   lb0 = __shfl(lab, k0), lb1 = __shfl(lab, k1);
            const float a0  = __shfl(xa,  k0), a1  = __shfl(xa,  k1);
            const float bp0 = __shfl(xbp, k0), bp1 = __shfl(xbp, k1);
            const float bg0 = __shfl(xbg, k0), bg1 = __shfl(xbg, k1);

            // A: one-hot of gt label 1..15 in rows 0..14, row 15 = all-ones
            v2f A;
            A.x = ((lb0 & 0xFFFF) == tgt) ? 1.0f : lastF;
            A.y = ((lb1 & 0xFFFF) == tgt) ? 1.0f : lastF;

            // Shared B selector (also the "ones" B matrix): 1.0 on the
            // one-hot col, lastF elsewhere (so col 15 is the wildcard).
            v2f Bo;
            Bo.x = ((lb0 >> 16) == tgt) ? 1.0f : lastF;
            Bo.y = ((lb1 >> 16) == tgt) ? 1.0f : lastF;

            // Value B matrices are just quantity * selector.
            v2f Ba, Bp, Bg;
            Ba.x = a0  * Bo.x;  Ba.y = a1  * Bo.y;
            Bp.x = bp0 * Bo.x;  Bp.y = bp1 * Bo.y;
            Bg.x = bg0 * Bo.x;  Bg.y = bg1 * Bo.y;

            acc[0] = wmma_f32_k4(A, Bo, acc[0]);
            acc[1] = wmma_f32_k4(A, Ba, acc[1]);
            acc[2] = wmma_f32_k4(A, Bp, acc[2]);
            acc[3] = wmma_f32_k4(A, Bg, acc[3]);
        }
    }

    // merge wave accumulators into block LDS (distinct addresses per lane ->
    // conflict-free within a wave, 8-way across waves)
    const int m_base = (lane >> 4) << 3;   // C/D layout: M = r + 8*(lane>=16)
    const int n_idx  = lane & 15;
#pragma unroll
    for (int q = 0; q < MATS; ++q) {
#pragma unroll
        for (int r = 0; r < 8; ++r) {
            atomicAdd(&lds[q * MAT_ELEMS + (m_base + r) * 16 + n_idx], acc[q][r]);
        }
    }
    __syncthreads();

    float* gp = partials + (long)img * ACC_PER_IMG;
    for (int t = threadIdx.x; t < ACC_PER_IMG; t += THREADS)
        atomicAdd(&gp[t], lds[t]);
}

// ---------------------------------------------------------------------------
// Tiny epilogue: IoU matching, dice, mean over images.  Marginals are read
// directly from the wildcard row (15) / col (15) of each matrix.
__global__ void finalize_kernel(const float* __restrict__ partials,
                                float* __restrict__ out) {
    __shared__ float sLoss[NIMG];
    const int t = threadIdx.x;

    if (t < NIMG) {
        const float* P  = partials + (long)t * ACC_PER_IMG;
        const float* Po = P + 0 * MAT_ELEMS;   // counts
        const float* Pa = P + 1 * MAT_ELEMS;   // A  = p*g*m
        const float* Pp = P + 2 * MAT_ELEMS;   // Bp = p*m
        const float* Pg = P + 3 * MAT_ELEMS;   // Bg = g*m
        const int MR = 15 * 16;                // wildcard (marginal) row offset

        float loss = 0.f;
        unsigned colm = 0u;
        for (int k = 1; k <= 15; ++k) {
            const int   kr    = (k - 1) * 16;
            const float areaG = Po[kr + 15];               // by_g counts
            const float gA    = Pa[kr + 15];
            const float gP    = Pp[kr + 15];
            const float gG    = Pg[kr + 15];
            float rowAcc = 0.f;
            bool  has    = false;
            for (int j = 1; j <= 15; ++j) {
                const int   e     = kr + (j - 1);
                const float apg   = Po[e];
                const float areaP = Po[MR + (j - 1)];      // by_p counts
                const float iou   = fmaxf(apg / areaP, apg / areaG);
                if (iou > IOU_T) {
                    has = true;
                    colm |= (1u << j);
                    const float inter = Pa[MR + (j - 1)] + gA - Pa[e];
                    const float psum  = Pp[MR + (j - 1)] + gP - Pp[e];
                    const float gsum  = Pg[MR + (j - 1)] + gG - Pg[e];
                    rowAcc += 1.f - 2.f * inter / (psum + gsum + EPS_F);
                }
            }
            if (has)
                loss += rowAcc;
            else
                loss += 1.f - 2.f * gA / (gP + gG + EPS_F);   // dice_gt
        }
        for (int j = 1; j <= 15; ++j)
            if (!(colm & (1u << j)))
                loss += 1.f - 2.f * Pa[MR + (j - 1)] /
                              (Pp[MR + (j - 1)] + Pg[MR + (j - 1)] + EPS_F);
        sLoss[t] = loss;
    }
    __syncthreads();

    if (t == 0) {
        float s = 0.f;
        for (int i = 0; i < NIMG; ++i) s += sLoss[i];
        out[0] = s / (float)NIMG;
    }
}

// ---------------------------------------------------------------------------
extern "C" void kernel_launch(void* const* d_in, const int* in_sizes, int n_in,
                              void* d_out, int out_size, void* d_ws, size_t ws_size,
                              hipStream_t stream) {
    const float* pred = (const float*)d_in[0];
    const float* gt   = (const float*)d_in[1];
    const float* mask = (const float*)d_in[2];
    const int*   plab = (const int*)d_in[3];
    const int*   glab = (const int*)d_in[4];
    float* partials = (float*)d_ws;                    // NIMG*4*256 floats = 32 KB
    float* outp     = (float*)d_out;

    const int nacc = NIMG * ACC_PER_IMG;
    zero_ws_kernel<<<(nacc + 255) / 256, 256, 0, stream>>>(partials, nacc);
    seg_wmma_kernel<<<NIMG * BLOCKS_PER_IMG, THREADS, 0, stream>>>(
        pred, gt, mask, plab, glab, partials);
    finalize_kernel<<<1, 32, 0, stream>>>(partials, outp);
}